// DSP_52183852646774
// MI455X (gfx1250) — compile-verified
//
#include <hip/hip_runtime.h>
#include <hip/hip_bf16.h>

typedef __attribute__((ext_vector_type(2))) float v2f;
typedef __attribute__((ext_vector_type(8))) float v8f;

#define BN_EPS 1e-3f

// ---------------------------------------------------------------------------
// K1: per-point stage-1 MLP (7->32, BN+ReLU folded) + atomic segment sums.
// ---------------------------------------------------------------------------
__global__ void __launch_bounds__(256) k1_accum(
    const float* __restrict__ pts, const float* __restrict__ fc,
    const int* __restrict__ inv, const float* __restrict__ W1,
    const float* __restrict__ g1, const float* __restrict__ b1,
    const float* __restrict__ m1, const float* __restrict__ v1,
    float* __restrict__ seg_sum, unsigned* __restrict__ counts, int N)
{
    __shared__ float sW1[224];
    __shared__ float sS[32], sT[32];
    int t = threadIdx.x;
    if (t < 224) sW1[t] = W1[t];
    if (t < 32) {
        float s = g1[t] * rsqrtf(v1[t] + BN_EPS);
        sS[t] = s;
        sT[t] = b1[t] - m1[t] * s;
    }
    __syncthreads();

    int i = blockIdx.x * 256 + t;
    if (i >= N) return;

    float f0 = fc[i * 3 + 0], f1 = fc[i * 3 + 1], f2 = fc[i * 3 + 2];
    float f3 = pts[i * 5 + 1], f4 = pts[i * 5 + 2];
    float f5 = pts[i * 5 + 3], f6 = pts[i * 5 + 4];

    int pid = inv[i];
    atomicAdd(&counts[pid], 1u);
    float* srow = seg_sum + (size_t)pid * 32;

#pragma unroll
    for (int c = 0; c < 32; ++c) {
        float acc = f0 * sW1[0 * 32 + c];
        acc = fmaf(f1, sW1[1 * 32 + c], acc);
        acc = fmaf(f2, sW1[2 * 32 + c], acc);
        acc = fmaf(f3, sW1[3 * 32 + c], acc);
        acc = fmaf(f4, sW1[4 * 32 + c], acc);
        acc = fmaf(f5, sW1[5 * 32 + c], acc);
        acc = fmaf(f6, sW1[6 * 32 + c], acc);
        float val = fmaxf(fmaf(acc, sS[c], sT[c]), 0.0f);
        // native global_atomic_add_f32 (relaxed, device scope)
        __hip_atomic_fetch_add(&srow[c], val, __ATOMIC_RELAXED,
                               __HIP_MEMORY_SCOPE_AGENT);
    }
}

// ---------------------------------------------------------------------------
// K2: seg_mean = seg_sum / max(count, 1)
// ---------------------------------------------------------------------------
__global__ void __launch_bounds__(256) k2_mean(
    const float* __restrict__ seg_sum, const unsigned* __restrict__ counts,
    float* __restrict__ seg_mean, int total)
{
    int i = blockIdx.x * 256 + threadIdx.x;
    if (i >= total) return;
    float cnt = (float)counts[i >> 5];
    seg_mean[i] = seg_sum[i] / fmaxf(cnt, 1.0f);
}

// ---------------------------------------------------------------------------
// K3: WMMA stage-2. Each wave handles 16 points:
//   A tile [16x64] = [h_row | seg_mean_row] staged in LDS (h recomputed),
//   B = W2 [64x32] in LDS, D = A*B via 16 x V_WMMA_F32_16X16X4_F32 per
//   16-col tile (2 tiles), then BN+ReLU and uint-bitpattern atomicMax.
// ---------------------------------------------------------------------------
__global__ void __launch_bounds__(256) k3_wmma(
    const float* __restrict__ pts, const float* __restrict__ fc,
    const int* __restrict__ inv,
    const float* __restrict__ W1,
    const float* __restrict__ g1, const float* __restrict__ b1,
    const float* __restrict__ m1, const float* __restrict__ v1,
    const float* __restrict__ seg_mean,
    const float* __restrict__ W2,
    const float* __restrict__ g2, const float* __restrict__ b2,
    const float* __restrict__ m2, const float* __restrict__ v2,
    unsigned* __restrict__ segmax, int N)
{
    __shared__ float sW1[224];
    __shared__ float sS1[32], sT1[32];
    __shared__ float sW2[64 * 32];
    __shared__ float sS2[32], sT2[32];
    __shared__ float sA[8][16 * 64];   // per-wave A tile (h2)
    __shared__ int   sPid[8][16];

    int t = threadIdx.x;
    if (t < 224) sW1[t] = W1[t];
#pragma unroll
    for (int j = t; j < 64 * 32; j += 256) sW2[j] = W2[j];
    if (t < 32) {
        float s1 = g1[t] * rsqrtf(v1[t] + BN_EPS);
        sS1[t] = s1; sT1[t] = b1[t] - m1[t] * s1;
        float s2 = g2[t] * rsqrtf(v2[t] + BN_EPS);
        sS2[t] = s2; sT2[t] = b2[t] - m2[t] * s2;
    }
    __syncthreads();

    int wave = t >> 5;
    int lane = t & 31;
    int p0 = blockIdx.x * 128 + wave * 16;
    float* A = &sA[wave][0];

    if (lane < 16) {
        // recompute stage-1 h row for point p0+lane -> A[row][0..31]
        int p = p0 + lane;
        int pc = (p < N) ? p : (N - 1);
        float f0 = fc[pc * 3 + 0], f1 = fc[pc * 3 + 1], f2 = fc[pc * 3 + 2];
        float f3 = pts[pc * 5 + 1], f4 = pts[pc * 5 + 2];
        float f5 = pts[pc * 5 + 3], f6 = pts[pc * 5 + 4];
        sPid[wave][lane] = inv[pc];
        float* Ar = A + lane * 64;
#pragma unroll
        for (int c = 0; c < 32; ++c) {
            float acc = f0 * sW1[0 * 32 + c];
            acc = fmaf(f1, sW1[1 * 32 + c], acc);
            acc = fmaf(f2, sW1[2 * 32 + c], acc);
            acc = fmaf(f3, sW1[3 * 32 + c], acc);
            acc = fmaf(f4, sW1[4 * 32 + c], acc);
            acc = fmaf(f5, sW1[5 * 32 + c], acc);
            acc = fmaf(f6, sW1[6 * 32 + c], acc);
            Ar[c] = fmaxf(fmaf(acc, sS1[c], sT1[c]), 0.0f);
        }
    } else {
        // gather seg_mean row -> A[row][32..63]
        int r = lane - 16;
        int p = p0 + r;
        int pc = (p < N) ? p : (N - 1);
        const float* mrow = seg_mean + (size_t)inv[pc] * 32;
        float* Ar = A + r * 64 + 32;
#pragma unroll
        for (int j = 0; j < 32; ++j) Ar[j] = mrow[j];
    }
    __syncthreads();

    // WMMA: D[16x32] = A[16x64] * W2[64x32], K split into 16 steps of 4.
    // A frag: lane L<16 holds M=L,(k0,k0+1); lane L>=16 holds M=L-16,(k0+2,k0+3)
    // B frag mirrors C/D striping: col = lane&15, K rows split by lane[4].
    v8f c0 = {};
    v8f c1 = {};
    int m = lane & 15;
    int kk = (lane >> 4) << 1;   // 0 or 2
    const float* Am = A + m * 64;
#pragma unroll
    for (int ks = 0; ks < 16; ++ks) {
        int k0 = ks * 4;
        v2f a;
        a.x = Am[k0 + kk];
        a.y = Am[k0 + kk + 1];
        const float* w0 = &sW2[(k0 + kk) * 32];
        const float* w1 = &sW2[(k0 + kk + 1) * 32];
        v2f bb0, bb1;
        bb0.x = w0[m];      bb0.y = w1[m];        // cols 0..15
        bb1.x = w0[16 + m]; bb1.y = w1[16 + m];   // cols 16..31
        c0 = __builtin_amdgcn_wmma_f32_16x16x4_f32(
            false, a, false, bb0, (short)0, c0, false, false);
        c1 = __builtin_amdgcn_wmma_f32_16x16x4_f32(
            false, a, false, bb1, (short)0, c1, false, false);
    }

    // C/D layout: row = (lane>>4)*8 + v, col = lane&15
    int nCol = lane & 15;
    int rbase = (lane >> 4) * 8;
    float s0 = sS2[nCol], t0 = sT2[nCol];
    float s1 = sS2[16 + nCol], t1 = sT2[16 + nCol];
#pragma unroll
    for (int v = 0; v < 8; ++v) {
        int row = rbase + v;
        int p = p0 + row;
        if (p < N) {
            int pid = sPid[wave][row];
            float x0 = fmaxf(fmaf(c0[v], s0, t0), 0.0f);
            float x1 = fmaxf(fmaf(c1[v], s1, t1), 0.0f);
            // ReLU output >= 0: uint bit-pattern order == float order
            atomicMax(&segmax[(size_t)pid * 32 + nCol], __float_as_uint(x0));
            atomicMax(&segmax[(size_t)pid * 32 + 16 + nCol], __float_as_uint(x1));
        }
    }
}

// ---------------------------------------------------------------------------
extern "C" void kernel_launch(void* const* d_in, const int* in_sizes, int n_in,
                              void* d_out, int out_size, void* d_ws, size_t ws_size,
                              hipStream_t stream)
{
    const float* pts = (const float*)d_in[0];   // [N,5]
    const float* fc  = (const float*)d_in[1];   // [N,3]
    const int*   inv = (const int*)d_in[2];     // [N]
    const float* W1  = (const float*)d_in[3];   // [7,32]
    const float* g1  = (const float*)d_in[4];
    const float* b1  = (const float*)d_in[5];
    const float* m1  = (const float*)d_in[6];
    const float* v1  = (const float*)d_in[7];
    const float* W2  = (const float*)d_in[8];   // [64,32]
    const float* g2  = (const float*)d_in[9];
    const float* b2  = (const float*)d_in[10];
    const float* m2  = (const float*)d_in[11];
    const float* v2  = (const float*)d_in[12];

    int N = in_sizes[0] / 5;
    int P = out_size / 32;

    // workspace: seg_sum [P*32 f32] | counts [P u32] | seg_mean [P*32 f32]
    float*    seg_sum  = (float*)d_ws;
    unsigned* counts   = (unsigned*)(seg_sum + (size_t)P * 32);
    float*    seg_mean = (float*)(counts + P);

    hipMemsetAsync(d_ws, 0, ((size_t)P * 32 + P) * sizeof(float), stream);
    hipMemsetAsync(d_out, 0, (size_t)P * 32 * sizeof(float), stream);

    k1_accum<<<(N + 255) / 256, 256, 0, stream>>>(
        pts, fc, inv, W1, g1, b1, m1, v1, seg_sum, counts, N);

    int total = P * 32;
    k2_mean<<<(total + 255) / 256, 256, 0, stream>>>(
        seg_sum, counts, seg_mean, total);

    k3_wmma<<<(N + 127) / 128, 256, 0, stream>>>(
        pts, fc, inv, W1, g1, b1, m1, v1, seg_mean, W2, g2, b2, m2, v2,
        (unsigned*)d_out, N);
}